// Dy_cross_Attention_90950227460848
// MI455X (gfx1250) — compile-verified
//
#include <hip/hip_runtime.h>
#include <math.h>

#define HH 512
#define WW 512
#define CC 32
#define HDIM 16
#define BB 2
#define NLI 50000
#define NRA 20000
#define ATT_SCALE 0.25f   // 1/sqrt(16)

// folded-constant block layout (floats)
#define CST_WQ   0
#define CST_WK   1024
#define CST_WV   2048
#define CST_BQ   3072
#define CST_BK   3104
#define CST_BV   3136
#define CST_KB   3168
#define CST_VB   3200
#define CST_PEV  3232
#define CST_SIZE 3520

typedef float v2f __attribute__((ext_vector_type(2)));
typedef float v8f __attribute__((ext_vector_type(8)));

__constant__ int SH9[9][2] = {{0,0},{-1,0},{1,0},{0,1},{-1,1},{1,1},{0,-1},{-1,-1},{1,-1}};

// ---------------- init kernels ----------------
__global__ void k_zero_f4(float4* p, long n4) {
  long i = (long)blockIdx.x * blockDim.x + threadIdx.x;
  if (i < n4) p[i] = make_float4(0.f, 0.f, 0.f, 0.f);
}

__global__ void k_fill_i32(int* p, long n, int v) {
  long i = (long)blockIdx.x * blockDim.x + threadIdx.x;
  if (i < n) p[i] = v;
}

// ---------------- grid build: grid[b][r][c] = pillar index ----------------
__global__ void k_build_grid(const int* __restrict__ coors, int* __restrict__ grid, int N) {
  int t = blockIdx.x * blockDim.x + threadIdx.x;
  int b = blockIdx.y;
  if (t >= N) return;
  const int* cp = coors + ((long)b * N + t) * 2;
  grid[(long)b * HH * WW + cp[0] * WW + cp[1]] = t;
}

// ---------------- weight folding (one block) ----------------
// WqF[n][k] = sum_c Ain[n][c]*qw[c][k] ; bqF[n] = sum_c Ain[n][c]*qb[c] + ainb[n]
// kb/vb     = raw in-proj biases (what invalid neighbors see, bias leak matches ref)
// pev[j][n] = sum_c pe9[j][c] * Ain_v[n][c]   (pos-enc pushed through Wv_in)
__global__ void k_fold(const float* __restrict__ qw, const float* __restrict__ qb,
                       const float* __restrict__ kw, const float* __restrict__ kb_,
                       const float* __restrict__ vw, const float* __restrict__ vb_,
                       const float* __restrict__ ainw, const float* __restrict__ ainb,
                       const float* __restrict__ posw, const float* __restrict__ posb,
                       float* __restrict__ cst) {
  int tid = threadIdx.x;
  for (int t = tid; t < 1024; t += 256) {
    int n = t >> 5, k = t & 31;
    float sq = 0.f, sk = 0.f, sv = 0.f;
    for (int c = 0; c < 32; c++) {
      sq += ainw[n * 32 + c]        * qw[c * 32 + k];
      sk += ainw[(32 + n) * 32 + c] * kw[c * 32 + k];
      sv += ainw[(64 + n) * 32 + c] * vw[c * 32 + k];
    }
    cst[CST_WQ + t] = sq;
    cst[CST_WK + t] = sk;
    cst[CST_WV + t] = sv;
  }
  for (int n = tid; n < 32; n += 256) {
    float sq = 0.f, sk = 0.f, sv = 0.f;
    for (int c = 0; c < 32; c++) {
      sq += ainw[n * 32 + c]        * qb[c];
      sk += ainw[(32 + n) * 32 + c] * kb_[c];
      sv += ainw[(64 + n) * 32 + c] * vb_[c];
    }
    cst[CST_BQ + n] = sq + ainb[n];
    cst[CST_BK + n] = sk + ainb[32 + n];
    cst[CST_BV + n] = sv + ainb[64 + n];
    cst[CST_KB + n] = ainb[32 + n];
    cst[CST_VB + n] = ainb[64 + n];
  }
  for (int t = tid; t < 288; t += 256) {
    int j = t >> 5, n = t & 31;
    float s0 = (float)SH9[j][0], s1 = (float)SH9[j][1];
    float acc = 0.f;
    for (int c = 0; c < 32; c++) {
      float pe = s0 * posw[c * 2] + s1 * posw[c * 2 + 1] + posb[c];
      acc += pe * ainw[(64 + n) * 32 + c];
    }
    cst[CST_PEV + t] = acc;
  }
}

// ---------------- LN + 3 projections via f32 WMMA ----------------
// one wave = 16 pillars; Y[m][n] = sum_k Xn[m][k]*W[n][k]  (A=X, B[k][n]=W[n][k])
// A frag (16x4 f32): lanes 0-15 = rows, x/y = K0/K1; lanes 16-31 x/y = K2/K3
// B frag (4x16 f32): lanes 0-15 = cols, x/y = K0/K1; lanes 16-31 x/y = K2/K3
__device__ __forceinline__ void proj_store(const v2f A[8], int lrow, int lh, int p0, int N,
                                           const float* __restrict__ Wm,
                                           const float* __restrict__ bv,
                                           float* __restrict__ op) {
#pragma unroll
  for (int nh = 0; nh < 2; nh++) {
    int n = nh * 16 + lrow;
    v8f acc = {0.f, 0.f, 0.f, 0.f, 0.f, 0.f, 0.f, 0.f};
#pragma unroll
    for (int kc = 0; kc < 8; kc++) {
      int k = kc * 4 + lh * 2;
      float2 bw = *(const float2*)(Wm + n * 32 + k);
      v2f Bf; Bf.x = bw.x; Bf.y = bw.y;
      acc = __builtin_amdgcn_wmma_f32_16x16x4_f32(false, A[kc], false, Bf,
                                                  (short)0, acc, false, false);
    }
    float bias = bv[n];
#pragma unroll
    for (int r = 0; r < 8; r++) {
      int p = p0 + r + lh * 8;
      if (p < N) op[(long)p * CC + n] = acc[r] + bias;
    }
  }
}

__global__ __launch_bounds__(256) void k_proj(
    const float* __restrict__ feats, int N,
    const float* __restrict__ lnw, const float* __restrict__ lnb,
    const float* __restrict__ W0, const float* __restrict__ b0, float* __restrict__ o0,
    const float* __restrict__ W1, const float* __restrict__ b1, float* __restrict__ o1,
    const float* __restrict__ W2, const float* __restrict__ b2, float* __restrict__ o2) {
  __shared__ float xt[8][16][33];
  int wv = threadIdx.x >> 5, lane = threadIdx.x & 31;
  int lrow = lane & 15, lh = lane >> 4;
  int b = blockIdx.y;
  int p0 = (blockIdx.x * 8 + wv) * 16;
  const float* fb = feats + (long)b * N * CC;

  for (int r = 0; r < 16; r++) {
    int p = p0 + r;
    xt[wv][r][lane] = (p < N) ? fb[(long)p * CC + lane] : 0.f;
  }
  __syncthreads();

  if (lh == 0) {  // LayerNorm, one lane per pillar row
    float mu = 0.f;
    for (int c = 0; c < 32; c++) mu += xt[wv][lrow][c];
    mu *= (1.f / 32.f);
    float var = 0.f;
    for (int c = 0; c < 32; c++) { float d = xt[wv][lrow][c] - mu; var += d * d; }
    var *= (1.f / 32.f);
    float inv = rsqrtf(var + 1e-5f);
    for (int c = 0; c < 32; c++)
      xt[wv][lrow][c] = (xt[wv][lrow][c] - mu) * inv * lnw[c] + lnb[c];
  }
  __syncthreads();

  v2f A[8];
#pragma unroll
  for (int kc = 0; kc < 8; kc++) {
    int k = kc * 4 + lh * 2;
    A[kc].x = xt[wv][lrow][k];
    A[kc].y = xt[wv][lrow][k + 1];
  }

  // fully unrolled: keep W0/W1/W2 as direct args so loads stay in global addrspace
  proj_store(A, lrow, lh, p0, N, W0, b0, o0 + (long)b * N * CC);
  proj_store(A, lrow, lh, p0, N, W1, b1, o1 + (long)b * N * CC);
  proj_store(A, lrow, lh, p0, N, W2, b2, o2 + (long)b * N * CC);
}

// ---------------- dy mask: 5x5 window mod 513 around dynamic radar pillars ----------------
__global__ void k_dymask(const unsigned char* __restrict__ dy, const int* __restrict__ ra_coors,
                         const int* __restrict__ li_grid, int* __restrict__ mask) {
  int t = blockIdx.x * blockDim.x + threadIdx.x;
  int b = blockIdx.y;
  if (t >= NRA) return;
  if (!dy[(long)b * NRA + t]) return;
  int r = ra_coors[((long)b * NRA + t) * 2];
  int c = ra_coors[((long)b * NRA + t) * 2 + 1];
  const int* g = li_grid + (long)b * HH * WW;
  for (int di = -2; di <= 2; di++)
    for (int dj = -2; dj <= 2; dj++) {
      int rr = r + di; if (rr < 0) rr += 513; else if (rr >= 513) rr -= 513;
      int cc = c + dj; if (cc < 0) cc += 513; else if (cc >= 513) cc -= 513;
      if (rr < HH && cc < WW) {
        int id = g[rr * WW + cc];
        if (id >= 0) mask[(long)b * NLI + id] = 1;
      }
    }
}

// ---------------- attention + WMMA out-projection + scatter ----------------
// lane = (pillar lrow, head lh); invalid neighbors see bias-only k/v (matches reference)
__global__ __launch_bounds__(256) void k_attn(
    const float* __restrict__ qh, const int* __restrict__ qcoors, int Nq,
    const float* __restrict__ kh, const float* __restrict__ vh, int Nkv,
    const int* __restrict__ grid_o, const float* __restrict__ cst,
    const float* __restrict__ Wo, const float* __restrict__ bo,
    const int* __restrict__ mask, float* __restrict__ outimg) {
  __shared__ float ot[8][16][33];
  __shared__ int crd[8][16][2];
  int wv = threadIdx.x >> 5, lane = threadIdx.x & 31;
  int lrow = lane & 15, lh = lane >> 4;  // lh = head during gather phase
  int b = blockIdx.y;
  int p0 = (blockIdx.x * 8 + wv) * 16;
  int p = p0 + lrow;
  bool act = p < Nq;
  int ps = act ? p : 0;  // clamped index for safe addressing

  int r = qcoors[((long)b * Nq + ps) * 2];
  int c = qcoors[((long)b * Nq + ps) * 2 + 1];
  if (lh == 0) { crd[wv][lrow][0] = r; crd[wv][lrow][1] = c; }

  float q[16];
  {
    const float* qr = qh + ((long)b * Nq + ps) * CC + lh * HDIM;
#pragma unroll
    for (int d = 0; d < 16; d++) q[d] = qr[d];
  }

  // bias-only key dot (hoisted): what every invalid neighbor scores
  float vb_reg[16];
  float dotb = 0.f;
  {
    const float* kbp = cst + CST_KB + lh * HDIM;
    const float* vbp = cst + CST_VB + lh * HDIM;
#pragma unroll
    for (int d = 0; d < 16; d++) {
      dotb += q[d] * kbp[d];
      vb_reg[d] = vbp[d];
    }
  }

  const int* g = grid_o + (long)b * HH * WW;
  float s[9]; int nid[9];
#pragma unroll
  for (int j = 0; j < 9; j++) {
    int rr = r + SH9[j][0], cc = c + SH9[j][1];
    int id = -1;
    if (act && rr >= 0 && rr < HH && cc >= 0 && cc < WW) id = g[rr * WW + cc];
    nid[j] = id;
    float dot = dotb;
    if (id >= 0) {  // address derives purely from kh -> global_load
      const float* kr = kh + ((long)b * Nkv + id) * CC + lh * HDIM;
      dot = 0.f;
#pragma unroll
      for (int d = 0; d < 16; d++) dot += q[d] * kr[d];
    }
    s[j] = dot * ATT_SCALE;
  }
  float mx = s[0];
#pragma unroll
  for (int j = 1; j < 9; j++) mx = fmaxf(mx, s[j]);
  float sum = 0.f;
#pragma unroll
  for (int j = 0; j < 9; j++) { s[j] = __expf(s[j] - mx); sum += s[j]; }
  float inv = 1.f / sum;

  float o[16];
#pragma unroll
  for (int d = 0; d < 16; d++) o[d] = 0.f;
  float ainv = 0.f;  // summed weight of invalid neighbors -> one bias contribution
#pragma unroll
  for (int j = 0; j < 9; j++) {
    float a = s[j] * inv;
    if (nid[j] >= 0) {
      const float* vr = vh + ((long)b * Nkv + nid[j]) * CC + lh * HDIM;
      const float* pr = cst + CST_PEV + j * CC + lh * HDIM;
#pragma unroll
      for (int d = 0; d < 16; d++) o[d] += a * (vr[d] + pr[d]);
    } else {
      ainv += a;
    }
  }
#pragma unroll
  for (int d = 0; d < 16; d++) o[d] += ainv * vb_reg[d];
  if (!act)
#pragma unroll
    for (int d = 0; d < 16; d++) o[d] = 0.f;
#pragma unroll
  for (int d = 0; d < 16; d++) ot[wv][lrow][lh * HDIM + d] = o[d];
  __syncthreads();

  // out-projection (32x32) via f32 WMMA, then masked scatter to [C,H,W]
  v2f A[8];
#pragma unroll
  for (int kc = 0; kc < 8; kc++) {
    int k = kc * 4 + lh * 2;
    A[kc].x = ot[wv][lrow][k];
    A[kc].y = ot[wv][lrow][k + 1];
  }
#pragma unroll
  for (int nh = 0; nh < 2; nh++) {
    int n = nh * 16 + lrow;
    v8f acc = {0.f, 0.f, 0.f, 0.f, 0.f, 0.f, 0.f, 0.f};
#pragma unroll
    for (int kc = 0; kc < 8; kc++) {
      int k = kc * 4 + lh * 2;
      float2 bw = *(const float2*)(Wo + n * 32 + k);
      v2f Bf; Bf.x = bw.x; Bf.y = bw.y;
      acc = __builtin_amdgcn_wmma_f32_16x16x4_f32(false, A[kc], false, Bf,
                                                  (short)0, acc, false, false);
    }
    float bias = bo[n];
#pragma unroll
    for (int rv = 0; rv < 8; rv++) {
      int m = rv + lh * 8;
      int pp = p0 + m;
      if (pp < Nq) {
        float val = acc[rv] + bias;
        if (mask) val *= (mask[(long)b * Nq + pp] ? 1.f : 0.f);
        int rr = crd[wv][m][0], cc = crd[wv][m][1];
        outimg[(((long)b * CC + n) * HH + rr) * WW + cc] = val;
      }
    }
  }
}

// ---------------- host ----------------
extern "C" void kernel_launch(void* const* d_in, const int* in_sizes, int n_in,
                              void* d_out, int out_size, void* d_ws, size_t ws_size,
                              hipStream_t stream) {
  const float* li_f = (const float*)d_in[0];
  const int*   li_c = (const int*)d_in[1];
  const float* ra_f = (const float*)d_in[2];
  const int*   ra_c = (const int*)d_in[3];
  const unsigned char* ra_dy = (const unsigned char*)d_in[4];
  const float* ln_li_w = (const float*)d_in[5];
  const float* ln_li_b = (const float*)d_in[6];
  const float* ln_ra_w = (const float*)d_in[7];
  const float* ln_ra_b = (const float*)d_in[8];
  const float* q1_w = (const float*)d_in[9];
  const float* q1_b = (const float*)d_in[10];
  const float* k1_w = (const float*)d_in[11];
  const float* k1_b = (const float*)d_in[12];
  const float* v1_w = (const float*)d_in[13];
  const float* v1_b = (const float*)d_in[14];
  const float* q2_w = (const float*)d_in[15];
  const float* q2_b = (const float*)d_in[16];
  const float* k2_w = (const float*)d_in[17];
  const float* k2_b = (const float*)d_in[18];
  const float* v2_w = (const float*)d_in[19];
  const float* v2_b = (const float*)d_in[20];
  const float* pos_w = (const float*)d_in[21];
  const float* pos_b = (const float*)d_in[22];
  const float* a1_in_w = (const float*)d_in[23];
  const float* a1_in_b = (const float*)d_in[24];
  const float* a1_out_w = (const float*)d_in[25];
  const float* a1_out_b = (const float*)d_in[26];
  const float* a2_in_w = (const float*)d_in[27];
  const float* a2_in_b = (const float*)d_in[28];
  const float* a2_out_w = (const float*)d_in[29];
  const float* a2_out_b = (const float*)d_in[30];

  char* ws = (char*)d_ws;
  size_t off = 0;
  auto take = [&](size_t bytes) -> char* {
    char* ptr = ws + off;
    off += (bytes + 255) & ~(size_t)255;
    return ptr;
  };
  int* li_grid = (int*)take((size_t)BB * HH * WW * 4);
  int* ra_grid = (int*)take((size_t)BB * HH * WW * 4);
  int* dymask  = (int*)take((size_t)BB * NLI * 4);
  float* qh1 = (float*)take((size_t)BB * NLI * CC * 4);
  float* kh1 = (float*)take((size_t)BB * NRA * CC * 4);
  float* vh1 = (float*)take((size_t)BB * NRA * CC * 4);
  float* qh2 = (float*)take((size_t)BB * NRA * CC * 4);
  float* kh2 = (float*)take((size_t)BB * NLI * CC * 4);
  float* vh2 = (float*)take((size_t)BB * NLI * CC * 4);
  float* cst1 = (float*)take(CST_SIZE * 4);
  float* cst2 = (float*)take(CST_SIZE * 4);

  float* out_li = (float*)d_out;
  float* out_ra = out_li + (size_t)BB * CC * HH * WW;

  // zero outputs + init grids/mask
  long n4 = (long)out_size / 4;
  k_zero_f4<<<dim3((unsigned)((n4 + 255) / 256)), 256, 0, stream>>>((float4*)d_out, n4);
  long ng = (long)BB * HH * WW;
  k_fill_i32<<<dim3((unsigned)((ng + 255) / 256)), 256, 0, stream>>>(li_grid, ng, -1);
  k_fill_i32<<<dim3((unsigned)((ng + 255) / 256)), 256, 0, stream>>>(ra_grid, ng, -1);
  long nm = (long)BB * NLI;
  k_fill_i32<<<dim3((unsigned)((nm + 255) / 256)), 256, 0, stream>>>(dymask, nm, 0);

  k_build_grid<<<dim3((NLI + 255) / 256, BB), 256, 0, stream>>>(li_c, li_grid, NLI);
  k_build_grid<<<dim3((NRA + 255) / 256, BB), 256, 0, stream>>>(ra_c, ra_grid, NRA);

  k_fold<<<1, 256, 0, stream>>>(q1_w, q1_b, k1_w, k1_b, v1_w, v1_b,
                                a1_in_w, a1_in_b, pos_w, pos_b, cst1);
  k_fold<<<1, 256, 0, stream>>>(q2_w, q2_b, k2_w, k2_b, v2_w, v2_b,
                                a2_in_w, a2_in_b, pos_w, pos_b, cst2);

  // lidar pillars feed: branch1 queries, branch2 keys/values
  k_proj<<<dim3((NLI + 127) / 128, BB), 256, 0, stream>>>(
      li_f, NLI, ln_li_w, ln_li_b,
      cst1 + CST_WQ, cst1 + CST_BQ, qh1,
      cst2 + CST_WK, cst2 + CST_BK, kh2,
      cst2 + CST_WV, cst2 + CST_BV, vh2);
  // radar pillars feed: branch2 queries, branch1 keys/values
  k_proj<<<dim3((NRA + 127) / 128, BB), 256, 0, stream>>>(
      ra_f, NRA, ln_ra_w, ln_ra_b,
      cst2 + CST_WQ, cst2 + CST_BQ, qh2,
      cst1 + CST_WK, cst1 + CST_BK, kh1,
      cst1 + CST_WV, cst1 + CST_BV, vh1);

  k_dymask<<<dim3((NRA + 255) / 256, BB), 256, 0, stream>>>(ra_dy, ra_c, li_grid, dymask);

  // branch 1: lidar queries x radar 3x3 neighbors, dy-masked
  k_attn<<<dim3((NLI + 127) / 128, BB), 256, 0, stream>>>(
      qh1, li_c, NLI, kh1, vh1, NRA, ra_grid, cst1, a1_out_w, a1_out_b, dymask, out_li);
  // branch 2: radar queries x lidar 3x3 neighbors, unmasked
  k_attn<<<dim3((NRA + 127) / 128, BB), 256, 0, stream>>>(
      qh2, ra_c, NRA, kh2, vh2, NLI, li_grid, cst2, a2_out_w, a2_out_b, nullptr, out_ra);
}